// HierarchicalSubsetBuilder_71451075937136
// MI455X (gfx1250) — compile-verified
//
#include <hip/hip_runtime.h>
#include <stdint.h>

// Problem constants (fixed by module config)
#define F_DIM 2048
#define TPB   256
#define EPT   (F_DIM / TPB)   // 8 elements per thread
#define KSB   4
#define NWAVE (TPB / 32)      // 8 wave32 per block

typedef uint32_t u32x4 __attribute__((ext_vector_type(4)));
typedef int      i32x4 __attribute__((ext_vector_type(4)));
typedef int      i32x8 __attribute__((ext_vector_type(8)));
typedef float    f32x4 __attribute__((ext_vector_type(4)));

// ---------------------------------------------------------------------------
// TDM: async DMA one row of F_DIM f32 from global memory into LDS.
// D# group0: count=1 (valid, load, user), lds_addr, 57-bit global addr, type=2.
// D# group1: data_size=4B, tensor_dim0=F, tensor_dim1=1, tile_dim0=F,
//            tile_dim1=1, tensor_dim0_stride=F. Groups 2/3 unused (<=2D).
// All operands are wave-uniform -> SGPR descriptor as the ISA requires.
// ---------------------------------------------------------------------------
__device__ __forceinline__ void tdm_load_row_to_lds(uint32_t lds_byte_off,
                                                    const float* gsrc) {
  uint64_t ga = (uint64_t)(uintptr_t)gsrc;
  u32x4 g0;
  g0.x = 1u;                                                // count=1, load
  g0.y = lds_byte_off;                                      // lds_addr (bytes)
  g0.z = (uint32_t)(ga & 0xFFFFFFFFull);                    // global_addr[31:0]
  g0.w = (uint32_t)((ga >> 32) & 0x1FFFFFFull) | (2u << 30);// addr[56:32]|type=2
  i32x8 g1;
  g1[0] = (int)(2u << 16);               // workgroup_mask=0, data_size=2 (4B)
  g1[1] = (int)((uint32_t)F_DIM << 16);  // tensor_dim0 lo16 at bit 48
  g1[2] = (int)(1u << 16);               // tensor_dim0 hi16=0, tensor_dim1=1
  g1[3] = (int)((uint32_t)F_DIM << 16);  // tensor_dim1 hi16=0, tile_dim0=F
  g1[4] = 1;                             // tile_dim1=1, tile_dim2=0
  g1[5] = F_DIM;                         // tensor_dim0_stride lo32
  g1[6] = 0;
  g1[7] = 0;
  i32x4 g2 = {0, 0, 0, 0};
  i32x4 g3 = {0, 0, 0, 0};
#if __has_include(<hip/amd_detail/amd_gfx1250_TDM.h>)
  // therock-10.0 headers present -> clang-23 6-arg builtin
  i32x8 g4 = {0, 0, 0, 0, 0, 0, 0, 0};
  __builtin_amdgcn_tensor_load_to_lds(g0, g1, g2, g3, g4, 0);
#else
  // ROCm 7.2 / clang-22 5-arg builtin
  __builtin_amdgcn_tensor_load_to_lds(g0, g1, g2, g3, 0);
#endif
}

// Intra-wave inclusive suffix scan (wave32, 5 shuffle steps, no barriers),
// fused over the 4 K-values. Lane l ends with sum of lanes [l..31].
__device__ __forceinline__ void wave_suffix_scan(uint32_t v[KSB], int lane) {
#pragma unroll
  for (int d = 1; d < 32; d <<= 1) {
#pragma unroll
    for (int k = 0; k < KSB; ++k) {
      uint32_t t = __shfl_down(v[k], (unsigned)d, 32);
      if (lane + d < 32) v[k] += t;
    }
  }
}

// ---------------------------------------------------------------------------
// One workgroup (256 threads, 8 wave32) processes rows in a grid-stride loop.
// Per row: keys = sortable-uint(scores + gumbel(u)); 4-pass radix select finds
// the exact K-th-largest key for K in {64,128,256,512} simultaneously; masks
// are key >= threshold, streamed out with non-temporal b128 stores.
// ---------------------------------------------------------------------------
__global__ __launch_bounds__(TPB) void hsb_topk_mask_kernel(
    const float* __restrict__ scores, const float* __restrict__ uin,
    float* __restrict__ out, int nrows) {
  __shared__ float    s_sc[2][F_DIM];     // 16 KB double-buffered scores
  __shared__ float    s_u [2][F_DIM];     // 16 KB double-buffered uniforms
  __shared__ uint32_t s_hist[KSB][TPB];   // 4 KB histograms (256 bins x 4 K)
  __shared__ uint32_t s_wtot[KSB][NWAVE]; // per-wave scan totals
  __shared__ uint32_t s_pref[KSB];        // selected key prefix per K
  __shared__ uint32_t s_rem [KSB];        // remaining rank within prefix

  const int  tid    = threadIdx.x;
  const int  lane   = tid & 31;
  const int  wv     = tid >> 5;
  const bool issuer = (wv == 0);          // wave 0 drives the TDM
  const int  stride = gridDim.x;
  const int  row0   = blockIdx.x;
  if (row0 >= nrows) return;

  // Prologue: async-load first row (2 TDM ops: scores + u)
  if (issuer) {
    tdm_load_row_to_lds((uint32_t)(uintptr_t)&s_sc[0][0],
                        scores + (size_t)row0 * F_DIM);
    tdm_load_row_to_lds((uint32_t)(uintptr_t)&s_u[0][0],
                        uin + (size_t)row0 * F_DIM);
  }

  int buf = 0;
  for (int row = row0; row < nrows; row += stride, buf ^= 1) {
    const int  nrow     = row + stride;
    const bool has_next = nrow < nrows;
    if (issuer) {
      if (has_next) {  // prefetch next row into other buffer, then wait for cur
        tdm_load_row_to_lds((uint32_t)(uintptr_t)&s_sc[buf ^ 1][0],
                            scores + (size_t)nrow * F_DIM);
        tdm_load_row_to_lds((uint32_t)(uintptr_t)&s_u[buf ^ 1][0],
                            uin + (size_t)nrow * F_DIM);
        __builtin_amdgcn_s_wait_tensorcnt(2);  // in-order: current row done
      } else {
        __builtin_amdgcn_s_wait_tensorcnt(0);
      }
    }
    __syncthreads();

    // ---- sortable keys for this thread's 8 contiguous elements ----
    uint32_t key[EPT];
    const f32x4* scv = (const f32x4*)&s_sc[buf][tid * EPT];
    const f32x4* uv  = (const f32x4*)&s_u [buf][tid * EPT];
#pragma unroll
    for (int q = 0; q < EPT / 4; ++q) {
      f32x4 sv = scv[q];
      f32x4 up = uv[q];
#pragma unroll
      for (int j = 0; j < 4; ++j) {
        float g = -__logf(-__logf(up[j] + 1e-20f) + 1e-20f);  // Gumbel
        float p = sv[j] + g;                                   // TAU = 1
        uint32_t b = __float_as_uint(p);
        // monotonic map: larger float -> larger uint
        key[q * 4 + j] = b ^ (uint32_t)((((int32_t)b) >> 31) | 0x80000000);
      }
    }

    if (tid < KSB) {
      const uint32_t KL[KSB] = {64u, 128u, 256u, 512u};
      s_pref[tid] = 0u;
      s_rem[tid]  = KL[tid];
    }

    // ---- 4-pass radix select, 8 bits per pass, 4 thresholds at once ----
#pragma unroll
    for (int pass = 0; pass < 4; ++pass) {
      const int shift = 24 - 8 * pass;
#pragma unroll
      for (int k = 0; k < KSB; ++k) s_hist[k][tid] = 0u;
      __syncthreads();

      uint32_t pref[KSB], rem[KSB];
#pragma unroll
      for (int k = 0; k < KSB; ++k) { pref[k] = s_pref[k]; rem[k] = s_rem[k]; }

      if (pass == 0) {  // all K share one histogram in the first pass
#pragma unroll
        for (int e = 0; e < EPT; ++e)
          atomicAdd(&s_hist[0][(key[e] >> shift) & 0xFF], 1u);
      } else {
#pragma unroll
        for (int e = 0; e < EPT; ++e) {
          uint32_t hi = key[e] >> (shift + 8);
          uint32_t lo = (key[e] >> shift) & 0xFF;
#pragma unroll
          for (int k = 0; k < KSB; ++k)
            if (hi == (pref[k] >> (shift + 8)))
              atomicAdd(&s_hist[k][lo], 1u);
        }
      }
      __syncthreads();

      // inclusive suffix sum over 256 bins: shuffle scan within each wave's
      // 32 bins, then add totals of higher-numbered waves (higher bins).
      uint32_t h[KSB], v[KSB];
#pragma unroll
      for (int k = 0; k < KSB; ++k) {
        h[k] = s_hist[pass == 0 ? 0 : k][tid];
        v[k] = h[k];
      }
      wave_suffix_scan(v, lane);
      if (lane == 0) {
#pragma unroll
        for (int k = 0; k < KSB; ++k) s_wtot[k][wv] = v[k];
      }
      __syncthreads();
#pragma unroll
      for (int k = 0; k < KSB; ++k) {
        uint32_t add = 0;
        for (int w2 = wv + 1; w2 < NWAVE; ++w2) add += s_wtot[k][w2];
        v[k] += add;
      }
      // unique winning bin per K: excl < rem <= incl
#pragma unroll
      for (int k = 0; k < KSB; ++k) {
        uint32_t excl = v[k] - h[k];  // count with byte > tid (same prefix)
        if (excl < rem[k] && rem[k] <= v[k]) {
          s_pref[k] = pref[k] | ((uint32_t)tid << shift);
          s_rem[k]  = rem[k] - excl;
        }
      }
      __syncthreads();
    }

    uint32_t thr[KSB];
#pragma unroll
    for (int k = 0; k < KSB; ++k) thr[k] = s_pref[k];  // exact K-th largest key

    // ---- stream hard masks: out[(row*4 + k)*F + f], NT b128 stores ----
#pragma unroll
    for (int k = 0; k < KSB; ++k) {
      float* obase = out + ((size_t)row * KSB + k) * F_DIM + tid * EPT;
#pragma unroll
      for (int q = 0; q < EPT / 4; ++q) {
        f32x4 m;
#pragma unroll
        for (int j = 0; j < 4; ++j)
          m[j] = (key[q * 4 + j] >= thr[k]) ? 1.0f : 0.0f;
        __builtin_nontemporal_store(m, (f32x4*)obase + q);
      }
    }
    __syncthreads();  // all reads of buf done before next TDM overwrites it
  }
}

extern "C" void kernel_launch(void* const* d_in, const int* in_sizes, int n_in,
                              void* d_out, int out_size, void* d_ws,
                              size_t ws_size, hipStream_t stream) {
  const float* scores = (const float*)d_in[0];
  const float* uin    = (const float*)d_in[1];
  float*       out    = (float*)d_out;
  const int nrows = in_sizes[0] / F_DIM;  // B*K_fc = 4096
  int grid = nrows < 1024 ? nrows : 1024; // 4 rows per WG -> TDM prefetch amortized
  if (grid < 1) grid = 1;
  hsb_topk_mask_kernel<<<grid, TPB, 0, stream>>>(scores, uin, out, nrows);
}